// NearestNeighbor_867583394193
// MI455X (gfx1250) — compile-verified
//
#include <hip/hip_runtime.h>

typedef float v2f __attribute__((ext_vector_type(2)));
typedef float v8f __attribute__((ext_vector_type(8)));

#define NN_B 4
#define NN_Q 8192
#define NN_P 8192

// One wave (32 threads) handles a 16-query tile against all NN_P points.
// s[q,p] = q·p - 0.5*||p||^2 computed by V_WMMA_F32_16X16X4_F32 with the
// 4th K slot carrying (1.0, -0.5*||p||^2); argmin(dist2) == argmax(s).
__global__ __launch_bounds__(32) void NearestNeighbor_wmma_kernel(
    const float* __restrict__ queries,   // [B, Q, 3]
    const float* __restrict__ points,    // [B, P, 3]
    int* __restrict__ out)               // [B, Q]
{
    const int lane = threadIdx.x;        // 0..31, wave32
    const int half = lane >> 4;          // 0 = lanes 0-15 (K=0,1), 1 = lanes 16-31 (K=2,3)
    const int sub  = lane & 15;          // row / column within tile
    const int qtiles = NN_Q / 16;
    const int b  = blockIdx.x / qtiles;
    const int q0 = (blockIdx.x % qtiles) * 16;

    // ---- Build A operand (16x4 f32, 2 VGPRs) ----
    // VGPR0: lanes 0-15 -> (M=sub, K=0)=qx ; lanes 16-31 -> (M=sub, K=2)=qz
    // VGPR1: lanes 0-15 -> (M=sub, K=1)=qy ; lanes 16-31 -> (M=sub, K=3)=1.0
    const size_t qbase = ((size_t)b * NN_Q + (size_t)(q0 + sub)) * 3;
    const float qx = queries[qbase + 0];
    const float qy = queries[qbase + 1];
    const float qz = queries[qbase + 2];
    v2f amat;
    amat.x = half ? qz : qx;
    amat.y = half ? 1.0f : qy;

    // Running argmax of s per accumulator row (8 rows per lane).
    float bestv[8];
    int   besti[8];
#pragma unroll
    for (int r = 0; r < 8; ++r) { bestv[r] = -__builtin_inff(); besti[r] = 0; }

    const size_t pbatch = (size_t)b * NN_P * 3;
    for (int pt = 0; pt < NN_P; pt += 16) {
        // ---- Build B operand (4x16 f32, 2 VGPRs) ----
        // VGPR0: lanes 0-15 -> (K=0, N=sub)=px ; lanes 16-31 -> (K=2, N=sub)=pz
        // VGPR1: lanes 0-15 -> (K=1, N=sub)=py ; lanes 16-31 -> (K=3, N=sub)=-0.5*psq
        const size_t pbase = pbatch + (size_t)(pt + sub) * 3;
        const float px = points[pbase + 0];
        const float py = points[pbase + 1];
        const float pz = points[pbase + 2];
        const float nhpsq = -0.5f * (px * px + py * py + pz * pz);
        v2f bmat;
        bmat.x = half ? pz : px;
        bmat.y = half ? nhpsq : py;

        v8f c = {};
        // 8 args: (neg_a, A, neg_b, B, c_mod, C, reuse_a, reuse_b)
        c = __builtin_amdgcn_wmma_f32_16x16x4_f32(
            false, amat, false, bmat, (short)0, c, false, false);

        // D layout: reg r, lane L -> row (r + 8*(L>=16)), col (L%16); point = pt + sub.
        const int pidx = pt + sub;
#pragma unroll
        for (int r = 0; r < 8; ++r) {
            const float s = c[r];
            const bool better = s > bestv[r];          // strict: keep earliest index on tie
            bestv[r] = better ? s : bestv[r];
            besti[r] = better ? pidx : besti[r];
        }
    }

    // ---- Reduce across the 16 lanes of each half (columns p%16) ----
    // Ties across columns resolved toward the smaller point index (argmin semantics).
#pragma unroll
    for (int r = 0; r < 8; ++r) {
        float bv = bestv[r];
        int   bi = besti[r];
#pragma unroll
        for (int s = 8; s >= 1; s >>= 1) {
            const float ov = __shfl_xor(bv, s, 32);
            const int   oi = __shfl_xor(bi, s, 32);
            const bool take = (ov > bv) || ((ov == bv) && (oi < bi));
            bv = take ? ov : bv;
            bi = take ? oi : bi;
        }
        bestv[r] = bv;
        besti[r] = bi;
    }

    // Lane 0 holds rows q0+0..7, lane 16 holds rows q0+8..15.
    if (sub == 0) {
        const int rowbase = q0 + (half ? 8 : 0);
#pragma unroll
        for (int r = 0; r < 8; ++r)
            out[(size_t)b * NN_Q + rowbase + r] = besti[r];
    }
}

extern "C" void kernel_launch(void* const* d_in, const int* in_sizes, int n_in,
                              void* d_out, int out_size, void* d_ws, size_t ws_size,
                              hipStream_t stream) {
    (void)in_sizes; (void)n_in; (void)d_ws; (void)ws_size; (void)out_size;
    const float* queries = (const float*)d_in[0];
    const float* points  = (const float*)d_in[1];
    int* out = (int*)d_out;

    const int blocks = NN_B * (NN_Q / 16);   // 2048 waves, one 16-query tile each
    NearestNeighbor_wmma_kernel<<<blocks, 32, 0, stream>>>(queries, points, out);
}